// FrameTransformerDecoder_47588237640240
// MI455X (gfx1250) — compile-verified
//
#include <hip/hip_runtime.h>
#include <math.h>

#define BB 2
#define CC 8
#define FF 256
#define WW 512
#define HH 8
#define EE 4
#define HDD 32
#define SZ (BB*CC*FF*WW)   // 2,097,152 floats = 8 MB per activation tensor

typedef __attribute__((ext_vector_type(2))) float v2f;
typedef __attribute__((ext_vector_type(8))) float v8f;

// ---------------------------------------------------------------------------
// fp32 WMMA: D = A(16x4) * B(4x16) + C(16x16), wave32, per CDNA5 ISA layouts.
// A: lanes 0-15 hold M=lane, VGPR0=K0 (VGPR1=K1); lanes 16-31 hold K2/K3.
// B: lanes 0-15 hold N=lane, VGPR0=K0 row; lanes 16-31 hold K2 row; VGPR1=K1/K3.
// C/D: VGPR r, lanes 0-15 -> M=r; lanes 16-31 -> M=r+8; N=lane&15.
// ---------------------------------------------------------------------------
__device__ __forceinline__ v8f wmma4(v2f a, v2f b, v8f c) {
  return __builtin_amdgcn_wmma_f32_16x16x4_f32(false, a, false, b, (short)0, c,
                                               false, false);
}

// ===========================================================================
// mc_linear:  Y[b,c,m,n] = sum_k Wm[c,m,k] * X[b,c,k,n]  (+ optional residual)
// One wave computes a 16(M) x 64(N) tile. grid = (N/64, M/16, B*C)
// ===========================================================================
__global__ __launch_bounds__(32) void gemm_kernel(
    const float* __restrict__ Aw, const float* __restrict__ Xin,
    const float* Rres, float* Yout, int M, int N, int K)
{
  const int bc    = blockIdx.z;
  const int c     = bc % CC;
  const int mbase = blockIdx.y * 16;
  const int nbase = blockIdx.x * 64;
  const float* Amat = Aw  + (size_t)c  * M * K;
  const float* Bmat = Xin + (size_t)bc * K * N;

  const int lane = threadIdx.x;
  const int lh   = lane & 15;
  const int ko   = (lane >> 4) << 1;          // 0 or 2 (K sub-offset per half-wave)

  v8f c0 = {}, c1 = {}, c2 = {}, c3 = {};
  const float* Ap = Amat + (size_t)(mbase + lh) * K + ko;
  const float* Bp = Bmat + (size_t)ko * N + nbase + lh;

  for (int k = 0; k < K; k += 4) {
    v2f av; av.x = Ap[k]; av.y = Ap[k + 1];
    const float* bp = Bp + (size_t)k * N;
    v2f b0, b1, b2, b3;
    b0.x = bp[0];   b0.y = bp[N];
    b1.x = bp[16];  b1.y = bp[N + 16];
    b2.x = bp[32];  b2.y = bp[N + 32];
    b3.x = bp[48];  b3.y = bp[N + 48];
    c0 = wmma4(av, b0, c0);
    c1 = wmma4(av, b1, c1);
    c2 = wmma4(av, b2, c2);
    c3 = wmma4(av, b3, c3);
  }

  float* Yp = Yout + (size_t)bc * M * N;
  const float* Rp = Rres ? Rres + (size_t)bc * M * N : nullptr;
  const int rowoff = (lane >> 4) << 3;
#pragma unroll
  for (int r = 0; r < 8; ++r) {
    int row = mbase + r + rowoff;
    size_t base = (size_t)row * N + nbase + lh;
    float r0 = c0[r], r1 = c1[r], r2 = c2[r], r3 = c3[r];
    if (Rp) { r0 += Rp[base]; r1 += Rp[base + 16]; r2 += Rp[base + 32]; r3 += Rp[base + 48]; }
    Yp[base]      = r0;
    Yp[base + 16] = r1;
    Yp[base + 32] = r2;
    Yp[base + 48] = r3;
  }
}

// ===========================================================================
// Attention core: one wave per (b,c,h, 16-query tile).
// S strip (16 x 512) via WMMA -> LDS, softmax in LDS, AV via WMMA.
// Output written transposed to (b,c, f=h*32+d, w=n) layout for Wo GEMM.
// ===========================================================================
__global__ __launch_bounds__(32) void attn_kernel(
    const float* __restrict__ Q, const float* __restrict__ Kt,
    const float* __restrict__ V, float* __restrict__ A2, float scale)
{
  __shared__ float S[16 * WW];                 // 32 KB

  const int blk   = blockIdx.x;
  const int ntile = blk & 31;                  // W/16 = 32 query tiles
  const int bch   = blk >> 5;                  // b*C*H + c*H + h
  const int bc    = bch / HH;
  const int hh    = bch % HH;

  const float* q = Q  + (size_t)bch * WW * HDD;
  const float* k = Kt + (size_t)bch * WW * HDD;
  const float* v = V  + (size_t)bch * WW * HDD;

  const int lane = threadIdx.x;
  const int lh   = lane & 15;
  const int ko   = (lane >> 4) << 1;
  const int rowoff = (lane >> 4) << 3;

  // Q fragments for all 8 K-steps (d = 0..31), reused across every m tile
  v2f aq[8];
  const float* qrow = q + (size_t)(ntile * 16 + lh) * HDD + ko;
#pragma unroll
  for (int s = 0; s < 8; ++s) { aq[s].x = qrow[4 * s]; aq[s].y = qrow[4 * s + 1]; }

  // ---- S = scale * Q K^T  (strip of 16 rows x 512 cols) ----
  for (int mt = 0; mt < WW / 16; ++mt) {
    v8f acc = {};
    const float* krow = k + (size_t)(mt * 16 + lh) * HDD + ko;
#pragma unroll
    for (int s = 0; s < 8; ++s) {
      v2f b; b.x = krow[4 * s]; b.y = krow[4 * s + 1];
      acc = wmma4(aq[s], b, acc);
    }
#pragma unroll
    for (int r = 0; r < 8; ++r)
      S[(r + rowoff) * WW + mt * 16 + lh] = acc[r] * scale;
  }
  __syncthreads();

  // ---- softmax over m (one lane per query row; LDS ops are in-order) ----
  if (lane < 16) {
    float* row = &S[lh * WW];
    float mx = -1e30f;
    for (int i = 0; i < WW; ++i) mx = fmaxf(mx, row[i]);
    float sum = 0.f;
    for (int i = 0; i < WW; ++i) { float e = __expf(row[i] - mx); row[i] = e; sum += e; }
    float inv = 1.f / sum;
    for (int i = 0; i < WW; ++i) row[i] *= inv;
  }
  __syncthreads();

  // ---- O = P V  (16 x 32, two N-halves) ----
  v8f o0 = {}, o1 = {};
  for (int mk = 0; mk < WW; mk += 4) {
    v2f a; a.x = S[lh * WW + mk + ko]; a.y = S[lh * WW + mk + ko + 1];
    const float* vb = v + (size_t)(mk + ko) * HDD + lh;
    v2f b0, b1;
    b0.x = vb[0];   b0.y = vb[HDD];
    b1.x = vb[16];  b1.y = vb[HDD + 16];
    o0 = wmma4(a, b0, o0);
    o1 = wmma4(a, b1, o1);
  }

  float* out = A2 + (size_t)bc * FF * WW;
#pragma unroll
  for (int r = 0; r < 8; ++r) {
    int n = ntile * 16 + r + rowoff;
    out[(size_t)(hh * HDD + lh) * WW + n]      = o0[r];
    out[(size_t)(hh * HDD + 16 + lh) * WW + n] = o1[r];
  }
}

// ===========================================================================
// MultichannelLayerNorm over F (256). One wave per (b,c,w) column.
// ===========================================================================
__global__ __launch_bounds__(256) void ln_kernel(
    const float* __restrict__ X, const float* __restrict__ g,
    const float* __restrict__ b, float* __restrict__ Y)
{
  const int wave = threadIdx.x >> 5;
  const int lane = threadIdx.x & 31;
  const int col  = blockIdx.x * 8 + wave;      // [0, B*C*W)
  const int w    = col % WW;
  const int bc   = col / WW;
  const int c    = bc % CC;

  const float* x = X + (size_t)bc * FF * WW + w;
  float vals[8]; float s = 0.f, ss = 0.f;
#pragma unroll
  for (int i = 0; i < 8; ++i) {
    float t = x[(size_t)(i * 32 + lane) * WW];
    vals[i] = t; s += t; ss += t * t;
  }
#pragma unroll
  for (int off = 16; off; off >>= 1) { s += __shfl_xor(s, off); ss += __shfl_xor(ss, off); }
  float mean = s * (1.f / FF);
  float var  = ss * (1.f / FF) - mean * mean;
  float rinv = rsqrtf(var + 1e-5f);

  float* y = Y + (size_t)bc * FF * WW + w;
#pragma unroll
  for (int i = 0; i < 8; ++i) {
    int f = i * 32 + lane;
    y[(size_t)f * WW] = (vals[i] - mean) * rinv * g[c * FF + f] + b[c * FF + f];
  }
}

// ===========================================================================
// Fused: 8x8x(1x3) channel-mix conv (pad_w=1) + head-split transpose + RoPE.
// in (b,c,f,w) -> out (b,c,h,w,hd). One thread per (b,co,h,pair j,w).
// ===========================================================================
__global__ __launch_bounds__(256) void convrope_kernel(
    const float* __restrict__ Yl, const float* __restrict__ Kc,
    const float* __restrict__ bias, float* __restrict__ Qo, int do_rope)
{
  int idx = blockIdx.x * blockDim.x + threadIdx.x;   // total = B*C*H*16*W
  const int w = idx % WW; int t = idx / WW;
  const int j = t % 16;  t /= 16;
  const int hh = t % HH; t /= HH;
  const int co = t % CC;
  const int b  = t / CC;

  const int f0 = hh * HDD + 2 * j;
  float acc0 = bias[co], acc1 = bias[co];
  for (int ci = 0; ci < CC; ++ci) {
    const float* src = Yl + (size_t)(b * CC + ci) * FF * WW;
    const float* kk  = Kc + (co * CC + ci) * 3;
#pragma unroll
    for (int dw = 0; dw < 3; ++dw) {
      int ws = w + dw - 1;
      if (ws >= 0 && ws < WW) {
        acc0 += kk[dw] * src[(size_t)f0 * WW + ws];
        acc1 += kk[dw] * src[(size_t)(f0 + 1) * WW + ws];
      }
    }
  }
  if (do_rope) {
    float inv = __powf(10000.f, -(float)(2 * j) / (float)HDD);
    float fr  = (float)w * inv;
    float cs  = __cosf(fr), sn = __sinf(fr);
    float r0  = acc0 * cs - acc1 * sn;
    float r1  = acc1 * cs + acc0 * sn;
    acc0 = r0; acc1 = r1;
  }
  float* dst = Qo + ((size_t)(b * CC + co) * HH + hh) * WW * HDD + (size_t)w * HDD;
  dst[2 * j]     = acc0;
  dst[2 * j + 1] = acc1;
}

// ===========================================================================
// Depthwise conv along F (kernel ksz, pad) + optional sq-relu + optional add.
// ===========================================================================
__global__ __launch_bounds__(256) void convf_kernel(
    const float* __restrict__ X, const float* __restrict__ Wk,
    const float* __restrict__ bias, const float* addsrc, float* Y,
    int ksz, int pad, int do_sqrelu)
{
  int idx = blockIdx.x * blockDim.x + threadIdx.x;   // total = SZ
  const int w = idx % WW; int t = idx / WW;
  const int f = t % FF;
  const int bcc = t / FF;
  const int c = bcc % CC;
  float acc = bias[c];
  for (int tt = 0; tt < ksz; ++tt) {
    int fs = f + tt - pad;
    if (fs >= 0 && fs < FF)
      acc += Wk[c * ksz + tt] * X[((size_t)bcc * FF + fs) * WW + w];
  }
  if (do_sqrelu) { acc = fmaxf(acc, 0.f); acc *= acc; }
  if (addsrc) acc += addsrc[idx];
  Y[idx] = acc;
}

// ===========================================================================
// FFN: mid = sqrelu(w3 @ z)  (E=4);   h += w4 @ mid
// ===========================================================================
__global__ __launch_bounds__(256) void ffn1_kernel(
    const float* __restrict__ Z, const float* __restrict__ w3, float* __restrict__ Mid)
{
  int idx = blockIdx.x * blockDim.x + threadIdx.x;   // total = B*C*E*W
  const int w = idx % WW; int t = idx / WW;
  const int g = t % EE;
  const int bcc = t / EE;
  const int c = bcc % CC;
  const float* x  = Z + (size_t)bcc * FF * WW + w;
  const float* wr = w3 + ((size_t)c * EE + g) * FF;
  float acc = 0.f;
  for (int f = 0; f < FF; ++f) acc += wr[f] * x[(size_t)f * WW];
  acc = fmaxf(acc, 0.f); acc *= acc;
  Mid[idx] = acc;
}

__global__ __launch_bounds__(256) void ffn2_kernel(
    const float* __restrict__ Mid, const float* __restrict__ w4, float* H)
{
  int idx = blockIdx.x * blockDim.x + threadIdx.x;   // total = SZ
  const int w = idx % WW; int t = idx / WW;
  const int f = t % FF;
  const int bcc = t / FF;
  const int c = bcc % CC;
  float acc = 0.f;
#pragma unroll
  for (int e = 0; e < EE; ++e)
    acc += w4[((size_t)c * FF + f) * EE + e] * Mid[((size_t)bcc * EE + e) * WW + w];
  H[idx] += acc;
}

// ===========================================================================
extern "C" void kernel_launch(void* const* d_in, const int* in_sizes, int n_in,
                              void* d_out, int out_size, void* d_ws, size_t ws_size,
                              hipStream_t stream) {
  const float* x    = (const float*)d_in[0];
  const float* skip = (const float*)d_in[1];
  const float* Wq   = (const float*)d_in[2];
  const float* Kq   = (const float*)d_in[3];
  const float* bq   = (const float*)d_in[4];
  const float* Wk   = (const float*)d_in[5];
  const float* Kk   = (const float*)d_in[6];
  const float* bk   = (const float*)d_in[7];
  const float* Wv   = (const float*)d_in[8];
  const float* Kv   = (const float*)d_in[9];
  const float* bv   = (const float*)d_in[10];
  const float* Wo   = (const float*)d_in[11];
  const float* ng   = (const float*)d_in[12];
  const float* nb   = (const float*)d_in[13];
  const float* c1aw = (const float*)d_in[14];
  const float* c1ab = (const float*)d_in[15];
  const float* c1bw = (const float*)d_in[16];
  const float* c1bb = (const float*)d_in[17];
  const float* c2w  = (const float*)d_in[18];
  const float* c2b  = (const float*)d_in[19];
  const float* w3   = (const float*)d_in[20];
  const float* w4   = (const float*)d_in[21];
  float* out = (float*)d_out;
  float* ws  = (float*)d_ws;

  float* h   = ws;
  float* z   = ws + (size_t)1 * SZ;
  float* y1  = ws + (size_t)2 * SZ;
  float* q   = ws + (size_t)3 * SZ;
  float* kbuf= ws + (size_t)4 * SZ;
  float* v   = ws + (size_t)5 * SZ;
  float* a2b = ws + (size_t)6 * SZ;
  float* t1  = ws + (size_t)7 * SZ;
  float* t2  = ws + (size_t)8 * SZ;
  float* z2  = ws + (size_t)9 * SZ;
  float* mid = ws + (size_t)10 * SZ;

  const dim3 ggrid(WW / 64, FF / 16, BB * CC);
  auto ln = [&](const float* src, int i, float* dst) {
    ln_kernel<<<BB * CC * WW / 8, 256, 0, stream>>>(src, ng + (size_t)i * CC * FF,
                                                    nb + (size_t)i * CC * FF, dst);
  };
  auto gemm = [&](const float* Wm, const float* X, const float* R, float* Y) {
    gemm_kernel<<<ggrid, 32, 0, stream>>>(Wm, X, R, Y, FF, WW, FF);
  };
  auto attention = [&](int i, const float* tin, const float* mem) {
    const size_t wo = (size_t)i * CC * FF * FF, co = (size_t)i * CC * CC * 3;
    const int crg = (SZ / 2) / 256;
    gemm(Wq + wo, tin, nullptr, y1);
    convrope_kernel<<<crg, 256, 0, stream>>>(y1, Kq + co, bq + i * CC, q, 1);
    gemm(Wk + wo, mem, nullptr, y1);
    convrope_kernel<<<crg, 256, 0, stream>>>(y1, Kk + co, bk + i * CC, kbuf, 1);
    gemm(Wv + wo, mem, nullptr, y1);
    convrope_kernel<<<crg, 256, 0, stream>>>(y1, Kv + co, bv + i * CC, v, 0);
    attn_kernel<<<BB * CC * HH * (WW / 16), 32, 0, stream>>>(q, kbuf, v, a2b,
                                                             1.0f / 16.0f); // 1/sqrt(F)
    gemm(Wo + wo, a2b, h, h);                              // h += Wo @ attn
  };

  // h = x  (embed is identity)
  hipMemcpyAsync(h, x, (size_t)SZ * sizeof(float), hipMemcpyDeviceToDevice, stream);

  // block 1: h = h + att0(z,z) + att1(z,skip)
  ln(h, 0, z);
  attention(0, z, z);
  attention(1, z, skip);

  // block 2: conv stack
  ln(h, 1, z);
  convf_kernel<<<SZ / 256, 256, 0, stream>>>(z, c1aw, c1ab, nullptr, t1, 11, 5, 1);
  convf_kernel<<<SZ / 256, 256, 0, stream>>>(z, c1bw, c1bb, t1, t2, 7, 3, 0);
  ln(t2, 2, z2);
  convf_kernel<<<SZ / 256, 256, 0, stream>>>(z2, c2w, c2b, h, h, 7, 3, 0);

  // blocks 3-4: self-attn then skip-attn
  ln(h, 3, z);
  attention(2, z, z);
  ln(h, 4, z);
  attention(3, z, skip);

  // block 5: FFN
  ln(h, 5, z);
  ffn1_kernel<<<(BB * CC * EE * WW) / 256, 256, 0, stream>>>(z, w3, mid);
  ffn2_kernel<<<SZ / 256, 256, 0, stream>>>(mid, w4, h);

  // concat along channels: out = (x, h)
  const size_t half = (size_t)CC * FF * WW;
  for (int b = 0; b < BB; ++b) {
    hipMemcpyAsync(out + (size_t)b * 2 * half,        x + (size_t)b * half,
                   half * sizeof(float), hipMemcpyDeviceToDevice, stream);
    hipMemcpyAsync(out + (size_t)b * 2 * half + half, h + (size_t)b * half,
                   half * sizeof(float), hipMemcpyDeviceToDevice, stream);
  }
}